// Model_7997229105633
// MI455X (gfx1250) — compile-verified
//
#include <hip/hip_runtime.h>

typedef __attribute__((ext_vector_type(2))) float v2f;
typedef __attribute__((ext_vector_type(8))) float v8f;

#define S_LEN 4096
#define H_DIM 16
#define P_DIM 64
#define N_DIM 64
#define KC 64            // time steps staged per LDS chunk
#define LDSTR 72         // padded LDS row stride (floats): 2*72 % 64 == 16 -> half-wave banks disjoint
#define NTHREADS 512     // 16 waves of 32
#define TSTRIDE (H_DIM * P_DIM)  // element stride between consecutive t in X/B

// out[b,h,p,n] = sum_t exp(G_total - G[t]) * X[b,t,h,p] * B[b,t,h,n]
// (weighted batched GEMM; the reference's chunked scan collapses to this)
__global__ __launch_bounds__(NTHREADS)
void mamba2_final_state_kernel(const float* __restrict__ X,
                               const float* __restrict__ A,
                               const float* __restrict__ B,
                               float* __restrict__ out)
{
    __shared__ float w[S_LEN];              // 16 KB: per-t weights
    __shared__ float sXw[KC * LDSTR];       // 18 KB: weighted X chunk [t][p]
    __shared__ float sB [KC * LDSTR];       // 18 KB: B chunk          [t][n]
    __shared__ float scanbuf[NTHREADS];     //  2 KB

    const int tid = threadIdx.x;
    const int bh  = blockIdx.x;             // 0..255 = b*16 + h
    const int b   = bh >> 4;
    const int h   = bh & 15;

    // ---------- Phase 1: w[t] = exp(total - inclusive_prefix(A)[t]) ----------
    float pref[8];
    float run = 0.f;
    const int t0 = tid * 8;
    #pragma unroll
    for (int j = 0; j < 8; ++j) {
        float a = A[((size_t)(b * S_LEN + t0 + j)) * H_DIM + h];
        run += a;
        pref[j] = run;                       // local inclusive prefix
    }
    scanbuf[tid] = run;
    __syncthreads();
    // Hillis-Steele inclusive scan over 512 per-thread partial sums
    for (int off = 1; off < NTHREADS; off <<= 1) {
        float mine  = scanbuf[tid];
        float other = (tid >= off) ? scanbuf[tid - off] : 0.f;
        __syncthreads();
        scanbuf[tid] = mine + other;
        __syncthreads();
    }
    const float excl  = scanbuf[tid] - run;
    const float total = scanbuf[NTHREADS - 1];
    #pragma unroll
    for (int j = 0; j < 8; ++j)
        w[t0 + j] = __expf(total - (excl + pref[j]));
    __syncthreads();

    // ---------- Phase 2: out(64x64) = (w*X)^T @ B via f32 WMMA ----------
    const int wave = tid >> 5;               // 0..15 -> 4x4 grid of 16x16 tiles
    const int lane = tid & 31;
    const int mt   = (wave >> 2) * 16;       // p-tile offset
    const int nt   = (wave & 3) * 16;        // n-tile offset
    const int row  = lane & 15;              // M (or N) index within tile
    const int kb   = (lane >> 4) * 2;        // K base: lanes 0-15 -> K{0,1}, 16-31 -> K{2,3}

    // per-thread staging geometry (iteration-invariant)
    const int trow0 = tid >> 4;              // 0..31 local time row (r=0)
    const int trow1 = trow0 + 32;            // 32..63 local time row (r=1)
    const int c4    = (tid & 15) * 4;        // column within row, float4 granularity

    v8f acc = {};

    const size_t base = (((size_t)b * S_LEN) * H_DIM + h) * (size_t)P_DIM; // X[b,0,h,0]

    // -------- software pipeline: hold next chunk in registers --------
    float4 xreg0, xreg1, breg0, breg1;
    {
        size_t g0 = base + (size_t)trow0 * TSTRIDE + c4;
        size_t g1 = base + (size_t)trow1 * TSTRIDE + c4;
        xreg0 = *(const float4*)(X + g0);
        xreg1 = *(const float4*)(X + g1);
        breg0 = *(const float4*)(B + g0);
        breg1 = *(const float4*)(B + g1);
    }

    for (int tc = 0; tc < S_LEN; tc += KC) {
        __syncthreads();                     // all waves done reading LDS (prev chunk)

        // scale + store current chunk (s_wait_loadcnt lands here, after a full
        // chunk of WMMA work has covered the load latency)
        {
            float w0 = w[tc + trow0];
            float w1 = w[tc + trow1];
            float4 x0 = xreg0, x1 = xreg1;
            x0.x *= w0; x0.y *= w0; x0.z *= w0; x0.w *= w0;
            x1.x *= w1; x1.y *= w1; x1.z *= w1; x1.w *= w1;
            *(float4*)(&sXw[trow0 * LDSTR + c4]) = x0;
            *(float4*)(&sXw[trow1 * LDSTR + c4]) = x1;
            *(float4*)(&sB [trow0 * LDSTR + c4]) = breg0;
            *(float4*)(&sB [trow1 * LDSTR + c4]) = breg1;
        }
        __syncthreads();

        // issue next chunk's global loads now; they complete during the WMMAs
        if (tc + KC < S_LEN) {
            size_t g0 = base + (size_t)(tc + KC + trow0) * TSTRIDE + c4;
            size_t g1 = base + (size_t)(tc + KC + trow1) * TSTRIDE + c4;
            xreg0 = *(const float4*)(X + g0);
            xreg1 = *(const float4*)(X + g1);
            breg0 = *(const float4*)(B + g0);
            breg1 = *(const float4*)(B + g1);
        }

        // 16 WMMAs per chunk: D(16x16) += A(16x4) x B(4x16), K stepping by 4
        #pragma unroll
        for (int kk = 0; kk < KC; kk += 4) {
            const float* xr = &sXw[(kk + kb) * LDSTR + mt + row];
            const float* br = &sB [(kk + kb) * LDSTR + nt + row];
            v2f af, bf;
            af.x = xr[0];
            af.y = xr[LDSTR];                // next K row
            bf.x = br[0];
            bf.y = br[LDSTR];
            acc = __builtin_amdgcn_wmma_f32_16x16x4_f32(
                      /*neg_a=*/false, af, /*neg_b=*/false, bf,
                      /*c_mod=*/(short)0, acc, /*reuse_a=*/false, /*reuse_b=*/false);
        }
    }

    // ---------- store D: VGPR j -> M = j + (lane>>4)*8, N = lane&15 ----------
    const int mhalf = (lane >> 4) * 8;
    float* o = out + (size_t)bh * (P_DIM * N_DIM);
    #pragma unroll
    for (int j = 0; j < 8; ++j) {
        int m = mt + mhalf + j;
        int n = nt + row;
        o[m * N_DIM + n] = acc[j];
    }
}

extern "C" void kernel_launch(void* const* d_in, const int* in_sizes, int n_in,
                              void* d_out, int out_size, void* d_ws, size_t ws_size,
                              hipStream_t stream) {
    const float* X = (const float*)d_in[0];
    const float* A = (const float*)d_in[1];
    const float* B = (const float*)d_in[2];
    // d_in[3] (C) is unused: the reference discards the C-dependent term.
    float* out = (float*)d_out;

    const int n_bh = 16 * H_DIM;             // b * h = 256 workgroups
    mamba2_final_state_kernel<<<n_bh, NTHREADS, 0, stream>>>(X, A, B, out);
}